// EnsembleModel_51281909514802
// MI455X (gfx1250) — compile-verified
//
#include <hip/hip_runtime.h>
#include <hip/hip_bf16.h>

// Problem constants from the reference (bz=64, sl=1024, n_pos=50).
#define BZ 64
#define SL 1024
#define NPOS 50
#define NBINS (NPOS * NPOS)     // 2500
#define ALPHA 0.3f
#define ROWS_PER_BLOCK 64
#define BLOCK 256

typedef float v4f __attribute__((ext_vector_type(4)));
typedef int   v4i __attribute__((ext_vector_type(4)));

// ---------------------------------------------------------------------------
// Histogram kernel (placed first so the disasm snippet shows the async path).
// Block-privatized histogram of a_arc into 2500 bins:
//   - a_arc rows streamed via CDNA5 async global->LDS copies (ASYNCcnt),
//     double-buffered so the fetch of row r+1 overlaps the LDS atomics of
//     row r; th:TH_LOAD_NT since a_arc is read-once (256MB stream >> 192MB L2).
//   - per-block LDS histogram (ds_add_f32), single global flush at the end
//     (1024 blocks * 2500 bins = 2.56M global atomics vs 67M if done globally).
// ---------------------------------------------------------------------------
__global__ __launch_bounds__(BLOCK) void hist_kernel(
    const float* __restrict__ a,        // [BZ, SL, SL]
    const int*   __restrict__ adds,     // [BZ, SL]
    float*       __restrict__ ghist) {  // [NBINS]
    __shared__ float hist[NBINS];
    __shared__ float tile[2][SL];       // two 4KB row buffers

    const int tid  = threadIdx.x;
    const int row0 = blockIdx.x * ROWS_PER_BLOCK;

    // Kick off async copy of the first row into tile[0] (each lane copies 16B).
    {
        unsigned long long ga = (unsigned long long)(a + ((size_t)row0 << 10) + tid * 4);
        unsigned int       la = (unsigned int)(size_t)(&tile[0][tid * 4]);
        asm volatile("global_load_async_to_lds_b128 %0, %1, off th:TH_LOAD_NT"
                     :: "v"(la), "v"(ga) : "memory");
    }

    for (int t = tid; t < NBINS; t += BLOCK) hist[t] = 0.0f;
    __syncthreads();

    int prev_b = -1;
    v4i aj     = (v4i)0;

    for (int r = 0; r < ROWS_PER_BLOCK; ++r) {
        const int row = row0 + r;
        const int b   = row >> 10;       // row / SL
        const int i   = row & (SL - 1);  // row % SL

        // Prefetch next row into the other buffer, then wait for current row.
        if (r + 1 < ROWS_PER_BLOCK) {
            unsigned long long ga =
                (unsigned long long)(a + ((size_t)(row + 1) << 10) + tid * 4);
            unsigned int la = (unsigned int)(size_t)(&tile[(r + 1) & 1][tid * 4]);
            asm volatile("global_load_async_to_lds_b128 %0, %1, off th:TH_LOAD_NT"
                         :: "v"(la), "v"(ga) : "memory");
            asm volatile("s_wait_asynccnt 0x1" ::: "memory");  // current row done
        } else {
            asm volatile("s_wait_asynccnt 0x0" ::: "memory");
        }

        if (b != prev_b) {  // adds[b, j] for this lane's 4 columns (L0-resident)
            aj = *(const v4i*)(adds + (b << 10) + tid * 4);
            prev_b = b;
        }
        const int base = adds[(b << 10) + i] * NPOS;  // uniform scalar load

        const v4f v = *(const v4f*)(&tile[r & 1][tid * 4]);  // ds_load_b128
        atomicAdd(&hist[base + aj[0]], v[0]);   // ds_add_f32
        atomicAdd(&hist[base + aj[1]], v[1]);
        atomicAdd(&hist[base + aj[2]], v[2]);
        atomicAdd(&hist[base + aj[3]], v[3]);
    }

    __syncthreads();
    for (int t = tid; t < NBINS; t += BLOCK)
        atomicAdd(&ghist[t], hist[t]);        // global_atomic_add_f32
}

// ---------------------------------------------------------------------------
// Zero the global histogram accumulator (d_ws is poisoned by the harness and
// must be re-initialized on every call).
// ---------------------------------------------------------------------------
__global__ void zero_hist_kernel(float* __restrict__ ghist) {
    int t = blockIdx.x * BLOCK + threadIdx.x;
    if (t < NBINS) ghist[t] = 0.0f;
}

// ---------------------------------------------------------------------------
// Apply kernel: out = s_arc + ALPHA * sigmoid(hist)[pos[b,i]*NPOS + pos[b,j]]
//   - sigmoid fused into the per-block LDS staging of the score table.
//   - non-temporal b128 streaming of s_arc / out (read-once / write-once).
// ---------------------------------------------------------------------------
__global__ __launch_bounds__(BLOCK) void apply_kernel(
    const float* __restrict__ s,        // [BZ, SL, SL]
    const int*   __restrict__ pos,      // [BZ, SL]
    const float* __restrict__ ghist,    // [NBINS]
    float*       __restrict__ out) {    // [BZ, SL, SL]
    __shared__ float score[NBINS];

    const int tid = threadIdx.x;
    for (int t = tid; t < NBINS; t += BLOCK) {
        float h = ghist[t];
        score[t] = 1.0f / (1.0f + __expf(-h));   // sigmoid, fused here
    }
    __syncthreads();

    const int row0 = blockIdx.x * ROWS_PER_BLOCK;
    int prev_b = -1;
    v4i pj     = (v4i)0;

    for (int r = 0; r < ROWS_PER_BLOCK; ++r) {
        const int row = row0 + r;
        const int b   = row >> 10;
        const int i   = row & (SL - 1);

        if (b != prev_b) {
            pj = *(const v4i*)(pos + (b << 10) + tid * 4);
            prev_b = b;
        }
        const int base = pos[(b << 10) + i] * NPOS;  // uniform scalar load

        const size_t off = ((size_t)row << 10) + tid * 4;
        v4f v = __builtin_nontemporal_load((const v4f*)(s + off)); // b128 NT load
        v[0] += ALPHA * score[base + pj[0]];          // ds_load gathers
        v[1] += ALPHA * score[base + pj[1]];
        v[2] += ALPHA * score[base + pj[2]];
        v[3] += ALPHA * score[base + pj[3]];
        __builtin_nontemporal_store(v, (v4f*)(out + off));         // b128 NT store
    }
}

// ---------------------------------------------------------------------------
extern "C" void kernel_launch(void* const* d_in, const int* in_sizes, int n_in,
                              void* d_out, int out_size, void* d_ws, size_t ws_size,
                              hipStream_t stream) {
    const float* a_arc = (const float*)d_in[0];
    const float* s_arc = (const float*)d_in[1];
    const int*   adds  = (const int*)d_in[2];
    const int*   pos   = (const int*)d_in[3];
    // d_in[4] is n_pos (==50), hardcoded as NPOS.

    float* ghist = (float*)d_ws;  // NBINS floats of scratch

    const int nblocks = (BZ * SL) / ROWS_PER_BLOCK;  // 1024

    zero_hist_kernel<<<(NBINS + BLOCK - 1) / BLOCK, BLOCK, 0, stream>>>(ghist);
    hist_kernel<<<nblocks, BLOCK, 0, stream>>>(a_arc, adds, ghist);
    apply_kernel<<<nblocks, BLOCK, 0, stream>>>(s_arc, pos, ghist, (float*)d_out);
}